// NMMONet_71657234367196
// MI455X (gfx1250) — compile-verified
//
#include <hip/hip_runtime.h>
#include <hip/hip_bf16.h>
#include <math.h>

// ---------------------------------------------------------------------------
// NMMONet forward: scatter_connection + LayerNormLSTM, tuned for gfx1250.
//   - scatter: LDS-accumulated (ds_add_f32), one (batch, channel-half) per WG
//   - input projection LN(x @ W_ih^T): hoisted, fused WMMA-f32 GEMM + row-LN
//   - recurrence: per step, WMMA-f32 GEMM (h @ W_hh^T) then fused
//     LN(2048) + gates + cell-LN(512) + h update, one WG per batch row
// ---------------------------------------------------------------------------

typedef __attribute__((ext_vector_type(2))) float v2f;
typedef __attribute__((ext_vector_type(8))) float v8f;

#define LN_EPS 1e-5f

__device__ __forceinline__ float sigm(float x) { return 1.0f / (1.0f + expf(-x)); }

// ============================ scatter_connection ============================
// grid (128, 2), block 256. Each WG owns batch b, channels [chunk*64, chunk*64+64).
__global__ __launch_bounds__(256)
void scatter_kernel(const float* __restrict__ emb,   // [128, 2048, 128]
                    const int*   __restrict__ loc,   // [128, 2048, 2]
                    float*       __restrict__ out)   // [128, 128, 15, 15]
{
    __shared__ float smap[64 * 225];                 // 56.25 KB LDS accumulator
    const int b     = blockIdx.x;
    const int chunk = blockIdx.y;

    for (int i = threadIdx.x; i < 64 * 225; i += 256) smap[i] = 0.0f;
    __syncthreads();

    const int ch  = threadIdx.x & 63;        // channel within chunk
    const int sub = threadIdx.x >> 6;        // 4 entities in flight
    const int chg = chunk * 64 + ch;

    for (int n = sub; n < 2048; n += 4) {
        const int base = (b * 2048 + n);
        const int y = loc[base * 2 + 0];
        const int x = loc[base * 2 + 1];
        const float v = emb[(size_t)base * 128 + chg];
        atomicAdd(&smap[ch * 225 + y * 15 + x], v);  // ds_add_f32
    }
    __syncthreads();

    float* ob = out + ((size_t)b * 128 + chunk * 64) * 225;
    for (int i = threadIdx.x; i < 64 * 225; i += 256) ob[i] = smap[i];
}

// ===================== Gx = LN(X @ W_ih^T) * g_i + b_i ======================
// X: [8192, 512], W_ih: [2048, 512] row-major (so out[m][n] = sum_k X[m][k]*W[n][k]).
// grid 512 (16 rows each), block 256 = 8 waves; wave w owns cols [w*256, w*256+256).
__global__ __launch_bounds__(256)
void gx_kernel(const float* __restrict__ X,
               const float* __restrict__ Wih,
               const float* __restrict__ g_i,
               const float* __restrict__ b_i,
               float*       __restrict__ Gx)        // [8192, 2048]
{
    const int row0   = blockIdx.x * 16;
    const int wave   = threadIdx.x >> 5;
    const int lane   = threadIdx.x & 31;
    const int half   = lane >> 4;                   // K-pair select
    const int l16    = lane & 15;
    const int n_base = wave * 256;

    v8f acc[16];
#pragma unroll
    for (int j = 0; j < 16; ++j) acc[j] = (v8f){0,0,0,0,0,0,0,0};

    const float* xrow = X + (size_t)(row0 + l16) * 512;

    for (int kk = 0; kk < 512; kk += 4) {
        const int ka = kk + 2 * half;
        v2f a; a.x = xrow[ka]; a.y = xrow[ka + 1];
#pragma unroll
        for (int j = 0; j < 16; ++j) {
            const int n = n_base + j * 16 + l16;
            const float* wrow = Wih + (size_t)n * 512 + ka;
            v2f bf; bf.x = wrow[0]; bf.y = wrow[1];
            acc[j] = __builtin_amdgcn_wmma_f32_16x16x4_f32(
                false, a, false, bf, (short)0, acc[j], false, false);
        }
    }

    // --- row LayerNorm over all 2048 cols (cross-wave via LDS) ---
    float psum[8], psq[8];
#pragma unroll
    for (int e = 0; e < 8; ++e) { psum[e] = 0.f; psq[e] = 0.f; }
#pragma unroll
    for (int j = 0; j < 16; ++j)
#pragma unroll
        for (int e = 0; e < 8; ++e) { float v = acc[j][e]; psum[e] += v; psq[e] += v * v; }

    for (int m = 1; m < 16; m <<= 1) {
#pragma unroll
        for (int e = 0; e < 8; ++e) {
            psum[e] += __shfl_xor(psum[e], m, 32);
            psq[e]  += __shfl_xor(psq[e],  m, 32);
        }
    }

    __shared__ float s_sum[16][8], s_sq[16][8], s_mu[16], s_rs[16];
    if (l16 == 0) {
#pragma unroll
        for (int e = 0; e < 8; ++e) {
            s_sum[e + 8 * half][wave] = psum[e];
            s_sq [e + 8 * half][wave] = psq[e];
        }
    }
    __syncthreads();
    if (threadIdx.x < 16) {
        float s = 0.f, q = 0.f;
        for (int w = 0; w < 8; ++w) { s += s_sum[threadIdx.x][w]; q += s_sq[threadIdx.x][w]; }
        const float mu = s * (1.0f / 2048.0f);
        s_mu[threadIdx.x] = mu;
        s_rs[threadIdx.x] = rsqrtf(q * (1.0f / 2048.0f) - mu * mu + LN_EPS);
    }
    __syncthreads();

#pragma unroll
    for (int j = 0; j < 16; ++j) {
        const int n = n_base + j * 16 + l16;
        const float gi = g_i[n], bi = b_i[n];
#pragma unroll
        for (int e = 0; e < 8; ++e) {
            const int m = e + 8 * half;
            Gx[(size_t)(row0 + m) * 2048 + n] = (acc[j][e] - s_mu[m]) * s_rs[m] * gi + bi;
        }
    }
}

// ========================= Gh_raw = h @ W_hh^T ==============================
// grid (16, 8), block 256 = 8 waves; each wave computes one 16x16 tile, K=512.
__global__ __launch_bounds__(256)
void gh_kernel(const float* __restrict__ Hprev,    // [128, 512]
               const float* __restrict__ Whh,      // [2048, 512]
               float*       __restrict__ Gh)       // [128, 2048] raw
{
    const int wave = threadIdx.x >> 5;
    const int lane = threadIdx.x & 31;
    const int half = lane >> 4;
    const int l16  = lane & 15;
    const int m0   = blockIdx.y * 16;
    const int n    = blockIdx.x * 128 + wave * 16 + l16;

    v8f acc = (v8f){0,0,0,0,0,0,0,0};
    const float* hrow = Hprev + (size_t)(m0 + l16) * 512;
    const float* wrow = Whh   + (size_t)n * 512;

    for (int kk = 0; kk < 512; kk += 4) {
        const int ka = kk + 2 * half;
        v2f a; a.x = hrow[ka]; a.y = hrow[ka + 1];
        v2f bf; bf.x = wrow[ka]; bf.y = wrow[ka + 1];
        acc = __builtin_amdgcn_wmma_f32_16x16x4_f32(
            false, a, false, bf, (short)0, acc, false, false);
    }
#pragma unroll
    for (int e = 0; e < 8; ++e)
        Gh[(size_t)(m0 + e + 8 * half) * 2048 + n] = acc[e];
}

// ================= per-row LN + gates + cell update =========================
__device__ __forceinline__ float blk_sum256(float v, float* sbuf /*[8]*/)
{
    for (int m = 16; m >= 1; m >>= 1) v += __shfl_xor(v, m, 32);
    const int wave = threadIdx.x >> 5;
    const int lane = threadIdx.x & 31;
    if (lane == 0) sbuf[wave] = v;
    __syncthreads();
    float r = 0.f;
#pragma unroll
    for (int w = 0; w < 8; ++w) r += sbuf[w];
    __syncthreads();
    return r;
}

// grid 128 (one WG per batch row), block 256.
__global__ __launch_bounds__(256)
void cell_kernel(const float* __restrict__ GhRaw,  // [128, 2048]
                 const float* __restrict__ GxT,    // [128, 2048] (step t slice)
                 const float* __restrict__ g_h, const float* __restrict__ b_h,
                 const float* __restrict__ g_c, const float* __restrict__ b_c,
                 float* __restrict__ c_buf,        // [128, 512] in/out
                 float* __restrict__ h_out,        // outputs[t]: [128, 512]
                 float* __restrict__ c_final,      // cT
                 int last)
{
    __shared__ float sg[2048];
    __shared__ float sred[8];
    const int b = blockIdx.x;
    const float* gh = GhRaw + (size_t)b * 2048;
    const float* gx = GxT   + (size_t)b * 2048;

    // LN over 2048 of h-projection, then add precomputed LN'd x-projection
    float vals[8], s = 0.f, q = 0.f;
#pragma unroll
    for (int i = 0; i < 8; ++i) {
        const float v = gh[threadIdx.x + i * 256];
        vals[i] = v; s += v; q += v * v;
    }
    s = blk_sum256(s, sred);
    q = blk_sum256(q, sred);
    const float mu = s * (1.0f / 2048.0f);
    const float rs = rsqrtf(q * (1.0f / 2048.0f) - mu * mu + LN_EPS);
#pragma unroll
    for (int i = 0; i < 8; ++i) {
        const int n = threadIdx.x + i * 256;
        sg[n] = (vals[i] - mu) * rs * g_h[n] + b_h[n] + gx[n];
    }
    __syncthreads();

    // gates -> c_new (pre-LN), stats over 512
    float cn[2], s2 = 0.f, q2 = 0.f;
#pragma unroll
    for (int i = 0; i < 2; ++i) {
        const int hd = threadIdx.x + i * 256;
        const float ig = sg[hd];
        const float fg = sg[512 + hd];
        const float gg = sg[1024 + hd];
        const float cv = sigm(fg) * c_buf[(size_t)b * 512 + hd] + sigm(ig) * tanhf(gg);
        cn[i] = cv; s2 += cv; q2 += cv * cv;
    }
    s2 = blk_sum256(s2, sred);
    q2 = blk_sum256(q2, sred);
    const float mu2 = s2 * (1.0f / 512.0f);
    const float rs2 = rsqrtf(q2 * (1.0f / 512.0f) - mu2 * mu2 + LN_EPS);
#pragma unroll
    for (int i = 0; i < 2; ++i) {
        const int hd = threadIdx.x + i * 256;
        const float cl = (cn[i] - mu2) * rs2 * g_c[hd] + b_c[hd];
        c_buf[(size_t)b * 512 + hd] = cl;                 // carried cell = LN'd cell
        const float h = sigm(sg[1536 + hd]) * tanhf(cl);
        h_out[(size_t)b * 512 + hd] = h;
        if (last) c_final[(size_t)b * 512 + hd] = cl;
    }
}

// ================================ launcher ==================================
extern "C" void kernel_launch(void* const* d_in, const int* in_sizes, int n_in,
                              void* d_out, int out_size, void* d_ws, size_t ws_size,
                              hipStream_t stream)
{
    const float* emb  = (const float*)d_in[0];   // project_embeddings [128,2048,128]
    const int*   loc  = (const int*)  d_in[1];   // entity_location   [128,2048,2]
    const float* X    = (const float*)d_in[2];   // lstm_input        [64,128,512]
    const float* h0   = (const float*)d_in[3];   // [128,512]
    const float* c0   = (const float*)d_in[4];   // [128,512]
    const float* w_ih = (const float*)d_in[5];   // [2048,512]
    const float* w_hh = (const float*)d_in[6];   // [2048,512]
    const float* g_i  = (const float*)d_in[7];
    const float* b_i  = (const float*)d_in[8];
    const float* g_h  = (const float*)d_in[9];
    const float* b_h  = (const float*)d_in[10];
    const float* g_c  = (const float*)d_in[11];
    const float* b_c  = (const float*)d_in[12];
    (void)in_sizes; (void)n_in; (void)out_size; (void)ws_size;

    float* out         = (float*)d_out;
    float* scatter_out = out;                        // 128*128*225 = 3,686,400
    float* outputs     = out + 3686400;              // 64*128*512  = 4,194,304
    float* cT          = out + 3686400 + 4194304;    // 128*512

    float* ws    = (float*)d_ws;
    float* Gx    = ws;                               // 8192*2048 = 16,777,216 f
    float* Gh    = ws + 16777216;                    // 128*2048
    float* c_buf = ws + 16777216 + 262144;           // 128*512

    // 1) scatter_connection (independent of LSTM)
    scatter_kernel<<<dim3(128, 2), 256, 0, stream>>>(emb, loc, scatter_out);

    // 2) hoisted input projection + LN  (8192 rows of 16 -> 512 WGs)
    gx_kernel<<<512, 256, 0, stream>>>(X, w_ih, g_i, b_i, Gx);

    // 3) init carried cell state
    hipMemcpyAsync(c_buf, c0, (size_t)128 * 512 * sizeof(float),
                   hipMemcpyDeviceToDevice, stream);

    // 4) recurrence: 64 steps of (WMMA GEMM, fused cell update)
    for (int t = 0; t < 64; ++t) {
        const float* hprev = (t == 0) ? h0 : (outputs + (size_t)(t - 1) * 65536);
        gh_kernel<<<dim3(16, 8), 256, 0, stream>>>(hprev, w_hh, Gh);
        cell_kernel<<<128, 256, 0, stream>>>(Gh, Gx + (size_t)t * 128 * 2048,
                                             g_h, b_h, g_c, b_c,
                                             c_buf, outputs + (size_t)t * 65536,
                                             cT, t == 63 ? 1 : 0);
    }
}